// loss_generator_9826885174019
// MI455X (gfx1250) — compile-verified
//
#include <hip/hip_runtime.h>
#include <stdint.h>

#define HW     256
#define ROWS   32
#define R_COEF 0.2025f   // (DT*C_WAVE/DX)^2 = (0.01*45)^2

// Async-copy one 256-float row (1 KB) from global into LDS using the CDNA5
// async data mover: 64 lanes x b128. Tracked by ASYNCcnt.
__device__ __forceinline__ void async_row_b128(const float* __restrict__ gsrc,
                                               float* ldst, int tid)
{
    if (tid < 64) {
        // low 32 bits of a generic LDS pointer == wave-relative LDS byte offset
        uint32_t lds_off = (uint32_t)(size_t)(ldst + tid * 4);
        const float* g = gsrc + tid * 4;
        asm volatile("global_load_async_to_lds_b128 %0, %1, off"
                     :: "v"(lds_off), "v"(g)
                     : "memory");
    }
}

__global__ __launch_bounds__(256) void wave_stencil_kernel(
    const float* __restrict__ u, float* __restrict__ out)
{
    // 16B alignment required: async b128 writes 16 bytes per lane into LDS
    __shared__ __align__(16) float sUn[4][HW]; // ring: rows i-1,i,i+1 (+prefetch) of u[t+1]
    __shared__ __align__(16) float sUm[2][HW]; // ring: row i (+prefetch) of u[t]

    const int tid = threadIdx.x;             // column j
    const int t   = blockIdx.x;              // output time index, 0..999
    const int i0  = blockIdx.y * ROWS;       // first output row of this chunk

    const float* un = u + ((size_t)(t + 1) << 16);   // 256*256 = 65536 per slab
    const float* um = u + ((size_t)t << 16);
    float*       o  = out + ((size_t)t << 16);

    // ---- preload: un rows i0-1, i0, i0+1 ; um row i0 ----
    {
        int r = i0 - 1;
        if (r >= 1 && r <= 254) async_row_b128(un + (r << 8), sUn[(r + 1) & 3], tid);
        else                    sUn[(r + 1) & 3][tid] = 0.0f;
        r = i0;
        if (r >= 1 && r <= 254) async_row_b128(un + (r << 8), sUn[(r + 1) & 3], tid);
        else                    sUn[(r + 1) & 3][tid] = 0.0f;
        r = i0 + 1;
        if (r >= 1 && r <= 254) async_row_b128(un + (r << 8), sUn[(r + 1) & 3], tid);
        else                    sUn[(r + 1) & 3][tid] = 0.0f;
        r = i0;
        if (r >= 1 && r <= 254) async_row_b128(um + (r << 8), sUm[r & 1], tid);
        else                    sUm[r & 1][tid] = 0.0f;
    }
    asm volatile("s_wait_asynccnt 0x0" ::: "memory");
    __syncthreads();

    for (int ii = 0; ii < ROWS; ++ii) {
        const int i = i0 + ii;

        // ---- software pipeline: issue async copies for the NEXT row while
        //      computing the current one (writes go to disjoint ring slots) ----
        if (ii + 1 < ROWS) {
            int r = i + 2;                                   // un row for iter ii+1
            if (r >= 1 && r <= 254) async_row_b128(un + (r << 8), sUn[(r + 1) & 3], tid);
            else                    sUn[(r + 1) & 3][tid] = 0.0f;
            r = i + 1;                                       // um row for iter ii+1
            if (r >= 1 && r <= 254) async_row_b128(um + (r << 8), sUm[r & 1], tid);
            else                    sUm[r & 1][tid] = 0.0f;
        }

        // ---- compute output row i from LDS ----
        float val = 0.0f;
        if (i >= 1 && i <= 254 && tid >= 1 && tid <= 254) {
            const float* cr = sUn[(i + 1) & 3];              // center row (un[i])
            float c  = cr[tid];
            float up = sUn[i & 3][tid];                      // un[i-1] (zeros if i==1)
            float dn = sUn[(i + 2) & 3][tid];                // un[i+1] (zeros if i==254)
            float lf = (tid == 1)   ? 0.0f : cr[tid - 1];    // boundary col masked
            float rt = (tid == 254) ? 0.0f : cr[tid + 1];
            float m  = sUm[i & 1][tid];                      // unm1[i]
            val = (2.0f - 4.0f * R_COEF) * c + R_COEF * (up + dn + lf + rt) - m;
        }
        // streaming output: nontemporal store, keep L2 for the reused input slabs
        __builtin_nontemporal_store(val, o + (i << 8) + tid);

        asm volatile("s_wait_asynccnt 0x0" ::: "memory");
        __syncthreads();
    }
}

extern "C" void kernel_launch(void* const* d_in, const int* in_sizes, int n_in,
                              void* d_out, int out_size, void* d_ws, size_t ws_size,
                              hipStream_t stream)
{
    const float* u   = (const float*)d_in[0];   // (1002,1,256,256) f32
    float*       out = (float*)d_out;           // (1000,1,256,256) f32
    // note: d_in[1] ('c') is unused by the reference formula

    const int ntime = in_sizes[0] / (HW * HW) - 2;   // 1000
    // blockIdx.x = t varies fastest so blocks sharing a slab (t and t+1) are
    // dispatched adjacently -> L2 collapses the un/um double-read of each slab.
    dim3 grid(ntime, HW / ROWS);                      // (1000, 8)
    wave_stencil_kernel<<<grid, 256, 0, stream>>>(u, out);
}